// MacroGCN_41970420418160
// MI455X (gfx1250) — compile-verified
//
#include <hip/hip_runtime.h>
#include <math.h>

// ---------------------------------------------------------------------------
// 2-layer GCN on gfx1250.
//   h  = X @ W1            (WMMA f32 16x16x4)
//   hs = h * dinv[row]     (fused in GEMM epilogue; agg initialized to hs -> self loop)
//   agg[dst] += hs[src]    (global_atomic_add_f32, L2-resident)
//   h1 = relu(dinv*agg+b1)
//   ... same for layer 2, then log_softmax over 16 labels.
// ---------------------------------------------------------------------------

typedef __attribute__((ext_vector_type(2))) float v2f;
typedef __attribute__((ext_vector_type(8))) float v8f;

#define FEATS 128
#define HID   64
#define LAB   16

// ---- degree / normalization ------------------------------------------------

__global__ void k_init_deg(int* __restrict__ deg, int n) {
    int i = blockIdx.x * blockDim.x + threadIdx.x;
    if (i < n) deg[i] = 1;                      // self loop contributes 1
}

__global__ void k_deg_scatter(const int* __restrict__ edges, int E,
                              int* __restrict__ deg) {
    int e = blockIdx.x * blockDim.x + threadIdx.x;
    if (e < E) atomicAdd(&deg[edges[E + e]], 1);   // dst = edges[1][e]
}

__global__ void k_dinv(const int* __restrict__ deg, float* __restrict__ dinv, int n) {
    int i = blockIdx.x * blockDim.x + threadIdx.x;
    if (i < n) dinv[i] = rsqrtf((float)deg[i]);
}

// ---- GEMM1: hs = (X @ W1) * dinv[row];  agg1 = hs (self-loop init) ---------
// One wave per 16-row tile, all 64 output columns (4 accumulators).

__global__ void k_gemm1(const float* __restrict__ X, const float* __restrict__ W,
                        const float* __restrict__ dinv,
                        float* __restrict__ hs, float* __restrict__ agg, int nTiles) {
    int tile = blockIdx.x * (blockDim.x >> 5) + (threadIdx.x >> 5);
    if (tile >= nTiles) return;                 // wave-uniform: EXEC stays all-1s
    int lane = threadIdx.x & 31;
    int mrow = lane & 15;                       // A row / B,C col within tile
    int ko   = (lane >> 4) << 1;                // K offset 0 or 2 (f32 A/B layout)
    int rowBase = tile << 4;

    v8f acc[4] = {};
    const float* Arow = X + (size_t)(rowBase + mrow) * FEATS;

    #pragma unroll 4
    for (int k = 0; k < FEATS; k += 4) {
        float2 av = *(const float2*)(Arow + k + ko);   // contiguous K pair, 8B aligned
        v2f a; a.x = av.x; a.y = av.y;
        #pragma unroll
        for (int t = 0; t < 4; ++t) {
            int col = (t << 4) + mrow;
            v2f b;
            b.x = W[(k + ko) * HID + col];
            b.y = W[(k + ko + 1) * HID + col];
            acc[t] = __builtin_amdgcn_wmma_f32_16x16x4_f32(
                false, a, false, b, (short)0, acc[t], false, false);
        }
    }

    int mBase = (lane >> 4) << 3;               // C/D: VGPR r -> M = r + 8*(lane>=16)
    #pragma unroll
    for (int r = 0; r < 8; ++r) {
        int row = rowBase + mBase + r;
        float s = dinv[row];
        #pragma unroll
        for (int t = 0; t < 4; ++t) {
            float v = acc[t][r] * s;
            size_t off = (size_t)row * HID + (t << 4) + mrow;
            hs[off]  = v;
            agg[off] = v;                       // self-loop contribution
        }
    }
}

// ---- edge scatter layer 1: warp per edge, 64 channels (2 per lane) ---------

__global__ void k_scatter1(const int* __restrict__ edges, int E,
                           const float* __restrict__ hs, float* __restrict__ agg) {
    int e = blockIdx.x * (blockDim.x >> 5) + (threadIdx.x >> 5);
    if (e >= E) return;
    int lane = threadIdx.x & 31;
    int s = edges[e];
    int d = edges[E + e];
    const float* srow = hs  + (size_t)s * HID;
    float*       drow = agg + (size_t)d * HID;
    atomicAdd(&drow[lane],      srow[lane]);
    atomicAdd(&drow[lane + 32], srow[lane + 32]);
}

// ---- epilogue 1: h1 = relu(dinv*agg + b1) ----------------------------------

__global__ void k_epi1(const float* __restrict__ agg, const float* __restrict__ dinv,
                       const float* __restrict__ b, float* __restrict__ h1,
                       size_t total) {
    size_t i = (size_t)blockIdx.x * blockDim.x + threadIdx.x;
    if (i >= total) return;
    int row = (int)(i >> 6);
    int c   = (int)(i & 63);
    float v = agg[i] * dinv[row] + b[c];
    h1[i] = v > 0.f ? v : 0.f;
}

// ---- GEMM2: gs = (H1 @ W2) * dinv[row];  agg2 = gs -------------------------

__global__ void k_gemm2(const float* __restrict__ H, const float* __restrict__ W,
                        const float* __restrict__ dinv,
                        float* __restrict__ gs, float* __restrict__ agg, int nTiles) {
    int tile = blockIdx.x * (blockDim.x >> 5) + (threadIdx.x >> 5);
    if (tile >= nTiles) return;
    int lane = threadIdx.x & 31;
    int mrow = lane & 15;
    int ko   = (lane >> 4) << 1;
    int rowBase = tile << 4;

    v8f acc = {};
    const float* Arow = H + (size_t)(rowBase + mrow) * HID;
    #pragma unroll
    for (int k = 0; k < HID; k += 4) {
        float2 av = *(const float2*)(Arow + k + ko);
        v2f a; a.x = av.x; a.y = av.y;
        v2f b;
        b.x = W[(k + ko) * LAB + mrow];
        b.y = W[(k + ko + 1) * LAB + mrow];
        acc = __builtin_amdgcn_wmma_f32_16x16x4_f32(
            false, a, false, b, (short)0, acc, false, false);
    }

    int mBase = (lane >> 4) << 3;
    #pragma unroll
    for (int r = 0; r < 8; ++r) {
        int row = rowBase + mBase + r;
        float v = acc[r] * dinv[row];
        size_t off = (size_t)row * LAB + mrow;
        gs[off]  = v;
        agg[off] = v;                           // self loop
    }
}

// ---- edge scatter layer 2: thread per (edge, label) ------------------------

__global__ void k_scatter2(const int* __restrict__ edges, int E,
                           const float* __restrict__ gs, float* __restrict__ agg) {
    long idx = (long)blockIdx.x * blockDim.x + threadIdx.x;
    int e = (int)(idx >> 4);
    if (e >= E) return;
    int c = (int)(idx & 15);
    int s = edges[e];
    int d = edges[E + e];
    atomicAdd(&agg[(size_t)d * LAB + c], gs[(size_t)s * LAB + c]);
}

// ---- epilogue 2: bias + log_softmax over 16 labels -------------------------

__global__ void k_softmax(const float* __restrict__ agg, const float* __restrict__ dinv,
                          const float* __restrict__ b, float* __restrict__ out, int n) {
    int i = blockIdx.x * blockDim.x + threadIdx.x;
    if (i >= n) return;
    float x[LAB];
    float dv = dinv[i];
    float mx = -INFINITY;
    #pragma unroll
    for (int c = 0; c < LAB; ++c) {
        x[c] = agg[(size_t)i * LAB + c] * dv + b[c];
        mx = fmaxf(mx, x[c]);
    }
    float s = 0.f;
    #pragma unroll
    for (int c = 0; c < LAB; ++c) s += expf(x[c] - mx);
    float ls = logf(s);
    #pragma unroll
    for (int c = 0; c < LAB; ++c) out[(size_t)i * LAB + c] = x[c] - mx - ls;
}

// ---------------------------------------------------------------------------

extern "C" void kernel_launch(void* const* d_in, const int* in_sizes, int n_in,
                              void* d_out, int out_size, void* d_ws, size_t ws_size,
                              hipStream_t stream) {
    const float* X  = (const float*)d_in[0];
    const int*   Eg = (const int*)d_in[1];
    const float* W1 = (const float*)d_in[2];
    const float* b1 = (const float*)d_in[3];
    const float* W2 = (const float*)d_in[4];
    const float* b2 = (const float*)d_in[5];

    const int N = in_sizes[0] / FEATS;          // 100000
    const int E = in_sizes[1] / 2;              // 1600000

    char* ws = (char*)d_ws;
    auto al = [](size_t x) { return (x + 255) & ~(size_t)255; };
    size_t off = 0;
    int*   deg  = (int*)(ws + off);   off = al(off + (size_t)N * 4);
    float* dinv = (float*)(ws + off); off = al(off + (size_t)N * 4);
    float* hs   = (float*)(ws + off); off = al(off + (size_t)N * HID * 4);  // also h1
    float* agg1 = (float*)(ws + off); off = al(off + (size_t)N * HID * 4);
    float* gs   = (float*)(ws + off); off = al(off + (size_t)N * LAB * 4);
    float* agg2 = (float*)(ws + off); off = al(off + (size_t)N * LAB * 4);

    const int nTiles = (N + 15) / 16;           // 6250 (N divisible by 16)

    k_init_deg   <<<(N + 255) / 256, 256, 0, stream>>>(deg, N);
    k_deg_scatter<<<(E + 255) / 256, 256, 0, stream>>>(Eg, E, deg);
    k_dinv       <<<(N + 255) / 256, 256, 0, stream>>>(deg, dinv, N);

    k_gemm1      <<<(nTiles + 3) / 4, 128, 0, stream>>>(X, W1, dinv, hs, agg1, nTiles);
    k_scatter1   <<<(E + 7) / 8, 256, 0, stream>>>(Eg, E, hs, agg1);
    size_t tot1 = (size_t)N * HID;
    k_epi1       <<<(int)((tot1 + 255) / 256), 256, 0, stream>>>(agg1, dinv, b1, hs, tot1);

    k_gemm2      <<<(nTiles + 3) / 4, 128, 0, stream>>>(hs, W2, dinv, gs, agg2, nTiles);
    long tot2 = (long)E * LAB;
    k_scatter2   <<<(int)((tot2 + 255) / 256), 256, 0, stream>>>(Eg, E, gs, agg2);

    k_softmax    <<<(N + 255) / 256, 256, 0, stream>>>(agg2, dinv, b2, (float*)d_out, N);
}